// FAGCNConv_936302871061
// MI455X (gfx1250) — compile-verified
//
#include <hip/hip_runtime.h>
#include <math.h>

#define CHANNELS 128
#define EPS_F 0.1f

typedef __attribute__((ext_vector_type(2))) float v2f;
typedef __attribute__((ext_vector_type(8))) float v8f;

// ---------------------------------------------------------------------------
// 1) zero the per-destination edge-count histogram
// ---------------------------------------------------------------------------
__global__ void fagcn_zero_counts(int* __restrict__ counts, int n) {
    int i = blockIdx.x * blockDim.x + threadIdx.x;
    if (i < n) counts[i] = 0;
}

// ---------------------------------------------------------------------------
// 2) histogram of edges per destination node
// ---------------------------------------------------------------------------
__global__ void fagcn_count(const int* __restrict__ col, int* __restrict__ counts, int e) {
    int i = blockIdx.x * blockDim.x + threadIdx.x;
    if (i < e) atomicAdd(&counts[col[i]], 1);
}

// ---------------------------------------------------------------------------
// 3) exclusive prefix scan over counts (single workgroup; N = 50000).
//    Writes offsets[0..n] and a working cursor copy for the fill pass.
// ---------------------------------------------------------------------------
__global__ void fagcn_scan(const int* __restrict__ counts,
                           int* __restrict__ offsets,
                           int* __restrict__ cursor, int n) {
    __shared__ int tmp[1024];
    const int tid = threadIdx.x;
    int base = 0;
    for (int start = 0; start < n; start += 1024) {
        int i = start + tid;
        int v = (i < n) ? counts[i] : 0;
        tmp[tid] = v;
        __syncthreads();
        for (int off = 1; off < 1024; off <<= 1) {
            int t = (tid >= off) ? tmp[tid - off] : 0;
            __syncthreads();
            tmp[tid] += t;
            __syncthreads();
        }
        int incl = tmp[tid];
        int excl = incl - v;
        if (i < n) {
            offsets[i] = base + excl;
            cursor[i]  = base + excl;
        }
        int chunk_total = tmp[1023];
        __syncthreads();
        base += chunk_total;
    }
    if (tid == 0) offsets[n] = base;
}

// ---------------------------------------------------------------------------
// 4) Per-node gate projections via V_WMMA_F32_16X16X4_F32 (branchless body).
//    One wave32 computes a 16-node tile of [a | b] = x_tile(16x128) * W(128x2),
//    W col0 = gate_w[0:128], col1 = gate_w[128:256].
//    A (16x4 f32): lanes 0-15 M=0..15 hold K=0(v0),K=1(v1); lanes 16-31 K=2,3.
//    B (4x16):     lanes 0-15 N=0..15 hold K=0(v0),K=1(v1); lanes 16-31 K=2,3.
//    C: element (M,N) -> VGPR (M&7), lane (M>>3)*16 + N.
// ---------------------------------------------------------------------------
__global__ void fagcn_gate_wmma(const float* __restrict__ x,
                                const float* __restrict__ gw,
                                float* __restrict__ anode,
                                float* __restrict__ bnode, int n) {
    const int lane = threadIdx.x & 31;
    const int wv   = (blockIdx.x * blockDim.x + threadIdx.x) >> 5;
    const int base = wv * 16;
    if (base >= n) return;               // wave-uniform: EXEC stays all-ones

    const int m  = lane & 15;            // A: row within tile; B: column N
    const int hg = lane >> 4;            // 0 -> K{0,1}, 1 -> K{2,3}

    // Clamp the load row for tail tiles: invalid C rows are never stored,
    // so their A contents don't matter -> no divergent guard on the loads.
    int node = base + m;
    if (node >= n) node = n - 1;
    const float* __restrict__ xrow = x + (size_t)node * CHANNELS + hg * 2;

    // Column mask instead of divergent B loads: cols 0/1 real, others zeroed.
    const float* __restrict__ gwl = gw + ((m == 1) ? CHANNELS : 0) + hg * 2;
    const float bmask = (m < 2) ? 1.0f : 0.0f;

    v8f c = {0.f, 0.f, 0.f, 0.f, 0.f, 0.f, 0.f, 0.f};

    #pragma unroll 8
    for (int kb = 0; kb < CHANNELS / 4; ++kb) {
        v2f av;
        av.x = xrow[kb * 4];
        av.y = xrow[kb * 4 + 1];
        v2f bv;
        bv.x = gwl[kb * 4] * bmask;
        bv.y = gwl[kb * 4 + 1] * bmask;
        c = __builtin_amdgcn_wmma_f32_16x16x4_f32(
                /*neg_a=*/false, av, /*neg_b=*/false, bv,
                /*c_mod=*/(short)0, c, /*reuse_a=*/false, /*reuse_b=*/false);
    }

    // Column 0 of C = a[node], column 1 = b[node].
    if (m == 0) {
        #pragma unroll
        for (int r = 0; r < 8; ++r) {
            int nd = base + hg * 8 + r;
            if (nd < n) anode[nd] = c[r];
        }
    } else if (m == 1) {
        #pragma unroll
        for (int r = 0; r < 8; ++r) {
            int nd = base + hg * 8 + r;
            if (nd < n) bnode[nd] = c[r];
        }
    }
}

// ---------------------------------------------------------------------------
// 5) Per-edge score + CSR fill.  tanh output is in [-1,1], so exp() needs no
//    segment-max shift (softmax ratios identical).
// ---------------------------------------------------------------------------
__global__ void fagcn_score_fill(const int* __restrict__ row,
                                 const int* __restrict__ col,
                                 const float* __restrict__ anode,
                                 const float* __restrict__ bnode,
                                 const float* __restrict__ gb,
                                 int* __restrict__ cursor,
                                 float* __restrict__ exs,
                                 int* __restrict__ rows_sorted, int e) {
    int i = blockIdx.x * blockDim.x + threadIdx.x;
    if (i >= e) return;
    int r = row[i];
    int c = col[i];
    float s  = tanhf(anode[r] + bnode[c] + gb[0]);
    float ex = expf(s);
    int pos = atomicAdd(&cursor[c], 1);
    exs[pos]         = ex;
    rows_sorted[pos] = r;
}

// ---------------------------------------------------------------------------
// 6) One wave32 per destination node.  Edge metadata is consumed in chunks of
//    32: coalesced lane loads of exs/rows_sorted, then per-edge broadcast via
//    __shfl.  32 lanes x float4 = 128 channels; each edge contributes one
//    coalesced 512B L2-resident row read of x.  Output written once, no atomics.
// ---------------------------------------------------------------------------
__global__ void fagcn_propagate(const float* __restrict__ x,
                                const int* __restrict__ offsets,
                                const float* __restrict__ exs,
                                const int* __restrict__ rows_sorted,
                                float* __restrict__ out, int n) {
    const int lane = threadIdx.x & 31;
    const int node = (blockIdx.x * blockDim.x + threadIdx.x) >> 5;
    if (node >= n) return;

    const int s0 = offsets[node];
    const int s1 = offsets[node + 1];

    float  ps  = 0.f;
    float4 acc = make_float4(0.f, 0.f, 0.f, 0.f);

    for (int j0 = s0; j0 < s1; j0 += 32) {
        float exl  = 0.f;
        int   rowl = 0;
        const int j = j0 + lane;
        if (j < s1) {
            exl  = exs[j];
            rowl = rows_sorted[j];
        }
        ps += exl;                         // fused softmax-denominator partial
        const int cnt = min(32, s1 - j0);
        for (int t = 0; t < cnt; ++t) {
            float w = __shfl(exl, t, 32);
            int   r = __shfl(rowl, t, 32);
            const float4* xr = (const float4*)(x + (size_t)r * CHANNELS);
            float4 v = xr[lane];
            acc.x = fmaf(v.x, w, acc.x);
            acc.y = fmaf(v.y, w, acc.y);
            acc.z = fmaf(v.z, w, acc.z);
            acc.w = fmaf(v.w, w, acc.w);
        }
    }

    #pragma unroll
    for (int o = 16; o > 0; o >>= 1) ps += __shfl_xor(ps, o, 32);

    const float inv = (s1 > s0) ? (1.0f - EPS_F) / ps : 0.0f;
    const float4* xs = (const float4*)(x + (size_t)node * CHANNELS);
    float4 xv = xs[lane];
    float4 o4;
    o4.x = fmaf(acc.x, inv, EPS_F * xv.x);
    o4.y = fmaf(acc.y, inv, EPS_F * xv.y);
    o4.z = fmaf(acc.z, inv, EPS_F * xv.z);
    o4.w = fmaf(acc.w, inv, EPS_F * xv.w);
    ((float4*)(out + (size_t)node * CHANNELS))[lane] = o4;
}

// ---------------------------------------------------------------------------
extern "C" void kernel_launch(void* const* d_in, const int* in_sizes, int n_in,
                              void* d_out, int out_size, void* d_ws, size_t ws_size,
                              hipStream_t stream) {
    const float* x  = (const float*)d_in[0];   // [N, 128]
    const int*   ei = (const int*)d_in[1];     // [2, E]
    const float* gw = (const float*)d_in[2];   // [256, 1]
    const float* gb = (const float*)d_in[3];   // [1]
    float* out = (float*)d_out;

    const int N = in_sizes[0] / CHANNELS;
    const int E = in_sizes[1] / 2;
    const int* row = ei;
    const int* col = ei + E;

    // workspace carve-out (256B-aligned slabs)
    char* p = (char*)d_ws;
    auto alloc = [&](size_t bytes) -> char* {
        char* r = p;
        p += (bytes + 255) & ~(size_t)255;
        return r;
    };
    float* anode       = (float*)alloc((size_t)N * 4);
    float* bnode       = (float*)alloc((size_t)N * 4);
    int*   counts      = (int*)  alloc((size_t)N * 4);
    int*   offsets     = (int*)  alloc((size_t)(N + 1) * 4);
    int*   cursor      = (int*)  alloc((size_t)N * 4);
    float* exs         = (float*)alloc((size_t)E * 4);
    int*   rows_sorted = (int*)  alloc((size_t)E * 4);
    (void)ws_size; (void)n_in; (void)out_size;

    fagcn_zero_counts<<<(N + 255) / 256, 256, 0, stream>>>(counts, N);
    fagcn_count<<<(E + 255) / 256, 256, 0, stream>>>(col, counts, E);
    fagcn_scan<<<1, 1024, 0, stream>>>(counts, offsets, cursor, N);
    fagcn_gate_wmma<<<(N + 127) / 128, 256, 0, stream>>>(x, gw, anode, bnode, N);
    fagcn_score_fill<<<(E + 255) / 256, 256, 0, stream>>>(row, col, anode, bnode, gb,
                                                          cursor, exs, rows_sorted, E);
    fagcn_propagate<<<(N + 7) / 8, 256, 0, stream>>>(x, offsets, exs, rows_sorted, out, N);
}